// OptimizingFCCDecoder_51926154609206
// MI455X (gfx1250) — compile-verified
//
#include <hip/hip_runtime.h>
#include <math.h>

// ---------------------------------------------------------------------------
// OptimizingFCCDecoder for MI455X (gfx1250, wave32).
// 16 problems per wave, duplicated across half-waves so that
// V_WMMA_F32_16X16X4_F32 operands can be formed without cross-lane shuffles.
// All dense eigenbasis transforms (U / U^T, 16x16 padded) run on WMMA.
// ---------------------------------------------------------------------------

typedef __attribute__((ext_vector_type(2))) float v2f;
typedef __attribute__((ext_vector_type(8))) float v8f;

#define NSTARTS 6
#define NSTEPS 25
#define LRC 0.08f
#define W6C 0.5f
#define B1C 0.9f
#define B2C 0.999f
#define ADAM_EPS_C 1e-8f
#define TWO_PI_F 6.28318530717958647692f

// LDS float offsets
#define P_U4Y_RE 0
#define P_U4Y_IM 256
#define P_U4X_RE 512
#define P_U4X_IM 768
#define P_U6Y_RE 1024
#define P_U6Y_IM 1280
#define P_U6X_RE 1536
#define P_U6X_IM 1792
#define P_M4Y 2048
#define P_M4X 2064
#define P_M6Y 2080
#define P_M6X 2096
#define P_WS4RE 2112
#define P_WS4IM 2128
#define P_WS6RE 2144
#define P_WS6IM 2160
#define LDS_TOT 2176

// acc += (TRANSP ? M^T : M) @ x, batched over the wave's 16 problems.
// A: constant matrix from LDS. B: per-lane vector (half-waves duplicate problems).
template <bool TRANSP>
__device__ __forceinline__ void mv_acc(const float* __restrict__ M,
                                       const float (&x)[16], float sgn,
                                       v8f& acc, int lane) {
  const bool hi = lane >= 16;
  const int r = lane & 15;
#pragma unroll
  for (int kc = 0; kc < 4; ++kc) {
    const int kb = kc * 4;
    const int k0 = kb + (hi ? 2 : 0);
    v2f A, B;
    if (TRANSP) {
      A.x = M[k0 * 16 + r];
      A.y = M[(k0 + 1) * 16 + r];
    } else {
      A.x = M[r * 16 + k0];
      A.y = M[r * 16 + k0 + 1];
    }
    B.x = sgn * (hi ? x[kb + 2] : x[kb + 0]);
    B.y = sgn * (hi ? x[kb + 3] : x[kb + 1]);
    acc = __builtin_amdgcn_wmma_f32_16x16x4_f32(false, A, false, B, (short)0,
                                                acc, false, false);
  }
}

// D layout: lane n holds comps 0..7, lane n+16 comps 8..15 (same problem).
// Exchange across half-waves so every lane holds the full 16-vector.
__device__ __forceinline__ void mv_finish(const v8f& acc, float (&y)[16],
                                          int lane) {
  const bool hi = lane >= 16;
#pragma unroll
  for (int rr = 0; rr < 8; ++rr) {
    float mine = acc[rr];
    float oth = __shfl_xor(mine, 16, 32);
    y[rr] = hi ? oth : mine;
    y[rr + 8] = hi ? mine : oth;
  }
}

// y = Re(U z) = Ure*zre - Uim*zim      (also adjoint of analysis)
__device__ __forceinline__ void csynth(const float* Mre, const float* Mim,
                                       const float (&xre)[16],
                                       const float (&xim)[16], float (&y)[16],
                                       int lane) {
  v8f acc;
#pragma unroll
  for (int i = 0; i < 8; ++i) acc[i] = 0.0f;
  mv_acc<false>(Mre, xre, 1.0f, acc, lane);
  mv_acc<false>(Mim, xim, -1.0f, acc, lane);
  mv_finish(acc, y, lane);
}

// w = U^H v : wre = Ure^T v ; wim = -Uim^T v   (also adjoint of synthesis)
__device__ __forceinline__ void canalysis(const float* Mre, const float* Mim,
                                          const float (&v)[16], float (&wre)[16],
                                          float (&wim)[16], int lane) {
  v8f a1;
#pragma unroll
  for (int i = 0; i < 8; ++i) a1[i] = 0.0f;
  mv_acc<true>(Mre, v, 1.0f, a1, lane);
  mv_finish(a1, wre, lane);
  v8f a2;
#pragma unroll
  for (int i = 0; i < 8; ++i) a2[i] = 0.0f;
  mv_acc<true>(Mim, v, -1.0f, a2, lane);
  mv_finish(a2, wim, lane);
}

// z = E(th) o w  with E_j = exp(i*m_j*th)
__device__ __forceinline__ void applyE(const float* mv, float th,
                                       const float (&wre)[16],
                                       const float (&wim)[16], float (&zre)[16],
                                       float (&zim)[16]) {
#pragma unroll
  for (int j = 0; j < 16; ++j) {
    float s, c;
    __sincosf(mv[j] * th, &s, &c);
    zre[j] = c * wre[j] - s * wim[j];
    zim[j] = s * wre[j] + c * wim[j];
  }
}

// adjoint of applyE + accumulate d(loss)/d(th) using saved z = E o w.
__device__ __forceinline__ void adjE(const float* mv, float th,
                                     const float (&zre)[16],
                                     const float (&zim)[16],
                                     const float (&gzre)[16],
                                     const float (&gzim)[16], float (&gwre)[16],
                                     float (&gwim)[16], float& gth) {
#pragma unroll
  for (int j = 0; j < 16; ++j) {
    float mj = mv[j];
    float s, c;
    __sincosf(mj * th, &s, &c);
    gth += mj * (gzim[j] * zre[j] - gzre[j] * zim[j]);
    gwre[j] = c * gzre[j] + s * gzim[j];
    gwim[j] = -s * gzre[j] + c * gzim[j];
  }
}

struct SHC {
  const float* Uyre;
  const float* Uyim;
  const float* Uxre;
  const float* Uxim;
  const float* my;
  const float* mx;
  const float* wsre;
  const float* wsim;
};

// Forward-only: returns weighted mean-squared loss for this l-channel.
__device__ __forceinline__ float sh_loss(const SHC& sh, float a, float b,
                                         float c, const float (&ft)[16],
                                         float dimf, float wgt, int lane) {
  float tre[16], tim[16];
#pragma unroll
  for (int j = 0; j < 16; ++j) {
    tre[j] = sh.wsre[j];
    tim[j] = sh.wsim[j];
  }
  float zcre[16], zcim[16];
  applyE(sh.my, c, tre, tim, zcre, zcim);
  float v1[16];
  csynth(sh.Uyre, sh.Uyim, zcre, zcim, v1, lane);
  float w1re[16], w1im[16];
  canalysis(sh.Uxre, sh.Uxim, v1, w1re, w1im, lane);
  float zbre[16], zbim[16];
  applyE(sh.mx, b, w1re, w1im, zbre, zbim);
  float v2[16];
  csynth(sh.Uxre, sh.Uxim, zbre, zbim, v2, lane);
  float w2re[16], w2im[16];
  canalysis(sh.Uyre, sh.Uyim, v2, w2re, w2im, lane);
  float zare[16], zaim[16];
  applyE(sh.my, a, w2re, w2im, zare, zaim);
  float f[16];
  csynth(sh.Uyre, sh.Uyim, zare, zaim, f, lane);
  float ls = 0.0f;
#pragma unroll
  for (int j = 0; j < 16; ++j) {
    float d = f[j] - ft[j];
    ls += d * d;
  }
  return wgt * ls / dimf;
}

// Forward + hand-derived reverse pass. Accumulates d(loss)/d(a,b,c).
__device__ __forceinline__ float sh_grad(const SHC& sh, float a, float b,
                                         float c, const float (&ft)[16],
                                         float dimf, float wgt, int lane,
                                         float& ga, float& gb, float& gc) {
  float tre[16], tim[16];
#pragma unroll
  for (int j = 0; j < 16; ++j) {
    tre[j] = sh.wsre[j];
    tim[j] = sh.wsim[j];
  }
  float zcre[16], zcim[16];
  applyE(sh.my, c, tre, tim, zcre, zcim);
  float v1[16];
  csynth(sh.Uyre, sh.Uyim, zcre, zcim, v1, lane);
  float w1re[16], w1im[16];
  canalysis(sh.Uxre, sh.Uxim, v1, w1re, w1im, lane);
  float zbre[16], zbim[16];
  applyE(sh.mx, b, w1re, w1im, zbre, zbim);
  float v2[16];
  csynth(sh.Uxre, sh.Uxim, zbre, zbim, v2, lane);
  float w2re[16], w2im[16];
  canalysis(sh.Uyre, sh.Uyim, v2, w2re, w2im, lane);
  float zare[16], zaim[16];
  applyE(sh.my, a, w2re, w2im, zare, zaim);
  float f[16];
  csynth(sh.Uyre, sh.Uyim, zare, zaim, f, lane);

  float ls = 0.0f;
  float gf[16];
  const float gscale = 2.0f * wgt / dimf;
#pragma unroll
  for (int j = 0; j < 16; ++j) {
    float d = f[j] - ft[j];
    ls += d * d;
    gf[j] = gscale * d;
  }

  // backward: adjoint chain (analysis<->synthesis swap)
  float gzre[16], gzim[16];
  canalysis(sh.Uyre, sh.Uyim, gf, gzre, gzim, lane);  // gz_a
  float gwre[16], gwim[16];
  adjE(sh.my, a, zare, zaim, gzre, gzim, gwre, gwim, ga);  // gw2, ga
  float gv[16];
  csynth(sh.Uyre, sh.Uyim, gwre, gwim, gv, lane);          // gv2
  canalysis(sh.Uxre, sh.Uxim, gv, gzre, gzim, lane);       // gz_b
  adjE(sh.mx, b, zbre, zbim, gzre, gzim, gwre, gwim, gb);  // gw1, gb
  csynth(sh.Uxre, sh.Uxim, gwre, gwim, gv, lane);          // gv1
  canalysis(sh.Uyre, sh.Uyim, gv, gzre, gzim, lane);       // gz_c
  adjE(sh.my, c, zcre, zcim, gzre, gzim, gwre, gwim, gc);  // gc
  return wgt * ls / dimf;
}

__device__ __forceinline__ float mod2pi(float x) {
  float r = fmodf(x, TWO_PI_F);
  return r + (r < 0.0f ? TWO_PI_F : 0.0f);
}

__global__ __launch_bounds__(256, 1) void fcc_adam_kernel(
    const float* __restrict__ f4, const float* __restrict__ f6,
    const float* __restrict__ s4, const float* __restrict__ s6,
    const float* __restrict__ q_init, const float* __restrict__ U4y,
    const float* __restrict__ lam4y, const float* __restrict__ U4x,
    const float* __restrict__ lam4x, const float* __restrict__ U6y,
    const float* __restrict__ lam6y, const float* __restrict__ U6x,
    const float* __restrict__ lam6x, float* __restrict__ wsq,
    float* __restrict__ wsl, int btot) {
  __shared__ float lds[LDS_TOT];
  const int tid = threadIdx.x;

  // ---- stage constants into LDS (zero-padded 16x16 planes) ----
  for (int t = tid; t < LDS_TOT; t += 256) lds[t] = 0.0f;
  __syncthreads();
  for (int t = tid; t < 81; t += 256) {
    int i = t / 9, j = t % 9;
    lds[P_U4Y_RE + i * 16 + j] = U4y[2 * t];
    lds[P_U4Y_IM + i * 16 + j] = U4y[2 * t + 1];
  }
  for (int t = tid; t < 81; t += 256) {
    int i = t / 9, j = t % 9;
    lds[P_U4X_RE + i * 16 + j] = U4x[2 * t];
    lds[P_U4X_IM + i * 16 + j] = U4x[2 * t + 1];
  }
  for (int t = tid; t < 169; t += 256) {
    int i = t / 13, j = t % 13;
    lds[P_U6Y_RE + i * 16 + j] = U6y[2 * t];
    lds[P_U6Y_IM + i * 16 + j] = U6y[2 * t + 1];
  }
  for (int t = tid; t < 169; t += 256) {
    int i = t / 13, j = t % 13;
    lds[P_U6X_RE + i * 16 + j] = U6x[2 * t];
    lds[P_U6X_IM + i * 16 + j] = U6x[2 * t + 1];
  }
  if (tid < 9) lds[P_M4Y + tid] = lam4y[2 * tid + 1];
  if (tid >= 16 && tid < 25) lds[P_M4X + tid - 16] = lam4x[2 * (tid - 16) + 1];
  if (tid >= 32 && tid < 45) lds[P_M6Y + tid - 32] = lam6y[2 * (tid - 32) + 1];
  if (tid >= 48 && tid < 61) lds[P_M6X + tid - 48] = lam6x[2 * (tid - 48) + 1];
  __syncthreads();
  // ws = Uy^H s (complex), per l
  if (tid < 16) {
    int j = tid;
    float ar = 0.0f, ai = 0.0f;
    for (int i = 0; i < 9; ++i) {
      float sv = s4[i];
      ar += lds[P_U4Y_RE + i * 16 + j] * sv;
      ai -= lds[P_U4Y_IM + i * 16 + j] * sv;
    }
    lds[P_WS4RE + j] = ar;
    lds[P_WS4IM + j] = ai;
  } else if (tid < 32) {
    int j = tid - 16;
    float ar = 0.0f, ai = 0.0f;
    for (int i = 0; i < 13; ++i) {
      float sv = s6[i];
      ar += lds[P_U6Y_RE + i * 16 + j] * sv;
      ai -= lds[P_U6Y_IM + i * 16 + j] * sv;
    }
    lds[P_WS6RE + j] = ar;
    lds[P_WS6IM + j] = ai;
  }
  __syncthreads();

  const SHC sh4{&lds[P_U4Y_RE], &lds[P_U4Y_IM], &lds[P_U4X_RE], &lds[P_U4X_IM],
                &lds[P_M4Y],    &lds[P_M4X],    &lds[P_WS4RE],  &lds[P_WS4IM]};
  const SHC sh6{&lds[P_U6Y_RE], &lds[P_U6Y_IM], &lds[P_U6X_RE], &lds[P_U6X_IM],
                &lds[P_M6Y],    &lds[P_M6X],    &lds[P_WS6RE],  &lds[P_WS6IM]};

  const int lane = tid & 31;
  const int wv = tid >> 5;
  int p = blockIdx.x * 128 + wv * 16 + (lane & 15);
  if (p >= btot) p = btot - 1;  // tail lanes duplicate (benign, deterministic)
  const int bb = p / NSTARTS;

  __builtin_prefetch(&f4[bb * 9], 0, 0);
  __builtin_prefetch(&f6[bb * 13], 0, 0);

  float ft4[16], ft6[16];
#pragma unroll
  for (int i = 0; i < 16; ++i) {
    ft4[i] = (i < 9) ? f4[bb * 9 + i] : 0.0f;
    ft6[i] = (i < 13) ? f6[bb * 13 + i] : 0.0f;
  }
  float u[4], am[4], av[4];
#pragma unroll
  for (int i = 0; i < 4; ++i) {
    u[i] = q_init[p * 4 + i];
    am[i] = 0.0f;
    av[i] = 0.0f;
  }
  const float invB = 1.0f / (float)btot;
  float pw1 = 1.0f, pw2 = 1.0f;

#pragma unroll 1
  for (int step = 0; step < NSTEPS; ++step) {
    pw1 *= B1C;
    pw2 *= B2C;
    // q = normalize(u)  (fix_sign irrelevant: R(-q)==R(q))
    float n2 = u[0] * u[0] + u[1] * u[1] + u[2] * u[2] + u[3] * u[3];
    float nn = sqrtf(n2);
    bool bad = (nn < 1e-12f);
    float inv = 1.0f / fmaxf(nn, 1e-12f);
    float qw = bad ? 1.0f : u[0] * inv;
    float qx = bad ? 0.0f : u[1] * inv;
    float qy = bad ? 0.0f : u[2] * inv;
    float qz = bad ? 0.0f : u[3] * inv;
    // R(q)
    float R00 = 1.0f - 2.0f * (qy * qy + qz * qz);
    float R01 = 2.0f * (qx * qy - qw * qz);
    float R02 = 2.0f * (qx * qz + qw * qy);
    float R11 = 1.0f - 2.0f * (qx * qx + qz * qz);
    float R20 = 2.0f * (qx * qz - qw * qy);
    float R21 = 2.0f * (qy * qz + qw * qx);
    float R22 = 1.0f - 2.0f * (qx * qx + qy * qy);
    // Euler extraction (matches _matrix_to_angles; Rab col0 = (cos a,0,-sin a))
    float x0 = R01, x1 = R11, x2 = R21;
    float nx = sqrtf(x0 * x0 + x1 * x1 + x2 * x2);
    float nxc = fmaxf(nx, 1e-12f);
    float xn0 = fminf(fmaxf(x0 / nxc, -1.0f), 1.0f);
    float xn1 = fminf(fmaxf(x1 / nxc, -1.0f), 1.0f);
    float xn2 = fminf(fmaxf(x2 / nxc, -1.0f), 1.0f);
    float bA = acosf(xn1);
    float aA = atan2f(xn0, xn2);
    float ca = cosf(aA), sa = sinf(aA);
    float Pc = ca * R02 - sa * R22;
    float Qc = ca * R00 - sa * R20;
    float cA = atan2f(Pc, Qc);
    float amod = mod2pi(aA), bmod = mod2pi(bA), cmod = mod2pi(cA);

    float ga = 0.0f, gb = 0.0f, gc = 0.0f;
    (void)sh_grad(sh4, amod, bmod, cmod, ft4, 9.0f, 1.0f, lane, ga, gb, gc);
    (void)sh_grad(sh6, amod, bmod, cmod, ft6, 13.0f, W6C, lane, ga, gb, gc);

    // backward: angles -> R
    float dc = Pc * Pc + Qc * Qc + 1e-20f;
    float gP = gc * Qc / dc, gQ = -gc * Pc / dc;
    float Pa = -sa * R02 - ca * R22;
    float Qa = -sa * R00 - ca * R20;
    ga += gP * Pa + gQ * Qa;
    float gR02 = gP * ca, gR22 = -gP * sa, gR00 = gQ * ca, gR20 = -gQ * sa;
    float da = xn0 * xn0 + xn2 * xn2 + 1e-20f;
    float gxn0 = ga * xn2 / da, gxn2 = -ga * xn0 / da;
    float sb = sqrtf(fmaxf(1.0f - xn1 * xn1, 1e-12f));
    float gxn1 = -gb / sb;
    float dotg = xn0 * gxn0 + xn1 * gxn1 + xn2 * gxn2;
    float gR01 = (gxn0 - xn0 * dotg) / nxc;
    float gR11 = (gxn1 - xn1 * dotg) / nxc;
    float gR21 = (gxn2 - xn2 * dotg) / nxc;
    // R -> q  (gR10 = gR12 = 0)
    float gq0 = 2.0f * (-qz * gR01 + qy * gR02 - qy * gR20 + qx * gR21);
    float gq1 = 2.0f * qy * gR01 + 2.0f * qz * gR02 - 4.0f * qx * gR11 +
                2.0f * qz * gR20 + 2.0f * qw * gR21 - 4.0f * qx * gR22;
    float gq2 = -4.0f * qy * gR00 + 2.0f * qx * gR01 + 2.0f * qw * gR02 -
                2.0f * qw * gR20 + 2.0f * qz * gR21 - 4.0f * qy * gR22;
    float gq3 = -4.0f * qz * gR00 - 2.0f * qw * gR01 + 2.0f * qx * gR02 -
                4.0f * qz * gR11 + 2.0f * qx * gR20 + 2.0f * qy * gR21;
    // q -> u (through normalize), plus mean over B
    float qdg = qw * gq0 + qx * gq1 + qy * gq2 + qz * gq3;
    float gu[4];
    gu[0] = bad ? 0.0f : (gq0 - qw * qdg) * inv * invB;
    gu[1] = bad ? 0.0f : (gq1 - qx * qdg) * inv * invB;
    gu[2] = bad ? 0.0f : (gq2 - qy * qdg) * inv * invB;
    gu[3] = bad ? 0.0f : (gq3 - qz * qdg) * inv * invB;
    // Adam
#pragma unroll
    for (int i = 0; i < 4; ++i) {
      am[i] = B1C * am[i] + (1.0f - B1C) * gu[i];
      av[i] = B2C * av[i] + (1.0f - B2C) * gu[i] * gu[i];
      float mh = am[i] / (1.0f - pw1);
      float vh = av[i] / (1.0f - pw2);
      u[i] -= LRC * mh / (sqrtf(vh) + ADAM_EPS_C);
    }
  }

  // final q = fix_sign(normalize(u)) and its loss
  float n2 = u[0] * u[0] + u[1] * u[1] + u[2] * u[2] + u[3] * u[3];
  float nn = sqrtf(n2);
  bool bad = (nn < 1e-12f);
  float inv = 1.0f / fmaxf(nn, 1e-12f);
  float qf[4];
  qf[0] = bad ? 1.0f : u[0] * inv;
  qf[1] = bad ? 0.0f : u[1] * inv;
  qf[2] = bad ? 0.0f : u[2] * inv;
  qf[3] = bad ? 0.0f : u[3] * inv;
  if (qf[0] < 0.0f) {
    qf[0] = -qf[0]; qf[1] = -qf[1]; qf[2] = -qf[2]; qf[3] = -qf[3];
  }
  {
    float qw = qf[0], qx = qf[1], qy = qf[2], qz = qf[3];
    float R00 = 1.0f - 2.0f * (qy * qy + qz * qz);
    float R01 = 2.0f * (qx * qy - qw * qz);
    float R02 = 2.0f * (qx * qz + qw * qy);
    float R11 = 1.0f - 2.0f * (qx * qx + qz * qz);
    float R20 = 2.0f * (qx * qz - qw * qy);
    float R21 = 2.0f * (qy * qz + qw * qx);
    float R22 = 1.0f - 2.0f * (qx * qx + qy * qy);
    float x0 = R01, x1 = R11, x2 = R21;
    float nx = sqrtf(x0 * x0 + x1 * x1 + x2 * x2);
    float nxc = fmaxf(nx, 1e-12f);
    float xn0 = fminf(fmaxf(x0 / nxc, -1.0f), 1.0f);
    float xn1 = fminf(fmaxf(x1 / nxc, -1.0f), 1.0f);
    float xn2 = fminf(fmaxf(x2 / nxc, -1.0f), 1.0f);
    float bA = acosf(xn1);
    float aA = atan2f(xn0, xn2);
    float ca = cosf(aA), sa = sinf(aA);
    float Pc = ca * R02 - sa * R22;
    float Qc = ca * R00 - sa * R20;
    float cA = atan2f(Pc, Qc);
    float amod = mod2pi(aA), bmod = mod2pi(bA), cmod = mod2pi(cA);
    float lv = sh_loss(sh4, amod, bmod, cmod, ft4, 9.0f, 1.0f, lane) +
               sh_loss(sh6, amod, bmod, cmod, ft6, 13.0f, W6C, lane);
    if (lane < 16) {
      wsl[p] = lv;
      wsq[p * 4 + 0] = qf[0];
      wsq[p * 4 + 1] = qf[1];
      wsq[p * 4 + 2] = qf[2];
      wsq[p * 4 + 3] = qf[3];
    }
  }
}

__global__ void fcc_select_kernel(const float* __restrict__ wsq,
                                  const float* __restrict__ wsl,
                                  float* __restrict__ out, int bsz) {
  int bb = blockIdx.x * blockDim.x + threadIdx.x;
  if (bb >= bsz) return;
  int best = 0;
  float bl = wsl[bb * NSTARTS];
#pragma unroll
  for (int k = 1; k < NSTARTS; ++k) {
    float l = wsl[bb * NSTARTS + k];
    if (l < bl) {
      bl = l;
      best = k;
    }
  }
  const float* q = &wsq[(bb * NSTARTS + best) * 4];
  float q0 = q[0], q1 = q[1], q2 = q[2], q3 = q[3];
  float nn = sqrtf(q0 * q0 + q1 * q1 + q2 * q2 + q3 * q3);
  bool bad = (nn < 1e-12f);
  float inv = 1.0f / fmaxf(nn, 1e-12f);
  q0 = bad ? 1.0f : q0 * inv;
  q1 = bad ? 0.0f : q1 * inv;
  q2 = bad ? 0.0f : q2 * inv;
  q3 = bad ? 0.0f : q3 * inv;
  if (q0 < 0.0f) {
    q0 = -q0; q1 = -q1; q2 = -q2; q3 = -q3;
  }
  out[bb * 4 + 0] = q0;
  out[bb * 4 + 1] = q1;
  out[bb * 4 + 2] = q2;
  out[bb * 4 + 3] = q3;
}

extern "C" void kernel_launch(void* const* d_in, const int* in_sizes, int n_in,
                              void* d_out, int out_size, void* d_ws,
                              size_t ws_size, hipStream_t stream) {
  const float* f4 = (const float*)d_in[0];
  const float* f6 = (const float*)d_in[1];
  const float* s4 = (const float*)d_in[2];
  const float* s6 = (const float*)d_in[3];
  const float* qi = (const float*)d_in[4];
  const float* U4y = (const float*)d_in[5];    // complex64 interleaved
  const float* lam4y = (const float*)d_in[6];  // complex64 interleaved
  const float* U4x = (const float*)d_in[7];
  const float* lam4x = (const float*)d_in[8];
  const float* U6y = (const float*)d_in[9];
  const float* lam6y = (const float*)d_in[10];
  const float* U6x = (const float*)d_in[11];
  const float* lam6x = (const float*)d_in[12];

  const int bsz = in_sizes[0] / 9;
  const int btot = bsz * NSTARTS;

  float* wsq = (float*)d_ws;             // btot*4 final quats
  float* wsl = wsq + (size_t)btot * 4;   // btot final losses

  const int blocks = (btot + 127) / 128;  // 128 problems per 256-thread block
  fcc_adam_kernel<<<blocks, 256, 0, stream>>>(f4, f6, s4, s6, qi, U4y, lam4y,
                                              U4x, lam4x, U6y, lam6y, U6x,
                                              lam6x, wsq, wsl, btot);
  fcc_select_kernel<<<(bsz + 255) / 256, 256, 0, stream>>>(wsq, wsl,
                                                           (float*)d_out, bsz);
}